// LinePooling_2748779070288
// MI455X (gfx1250) — compile-verified
//
#include <hip/hip_runtime.h>
#include <hip/hip_bf16.h>
#include <cstdint>

// ---------- compile-time probes (feedback via #warning in stderr) ----------
#if __has_builtin(__builtin_amdgcn_global_load_async_to_lds_b32)
#warning "PROBE: __builtin_amdgcn_global_load_async_to_lds_b32 AVAILABLE"
#endif
#if __has_builtin(__builtin_amdgcn_global_load_async_to_lds_b128)
#warning "PROBE: __builtin_amdgcn_global_load_async_to_lds_b128 AVAILABLE"
#endif
#if __has_builtin(__builtin_amdgcn_s_wait_asynccnt)
#warning "PROBE: __builtin_amdgcn_s_wait_asynccnt AVAILABLE"
#endif

// ---------------- problem constants ----------------
#define NSAMP   32
#define NFIN    8
#define CTOT    128
#define HIMG    128
#define WIMG    128
#define HW_     (HIMG * WIMG)          // 16384 pixels (2^14)
#define NLINES  50000

// ---------------- tiling ----------------
#define CB              4              // channels staged per workgroup (256 KB LDS)
#define NCB             (CTOT / CB)    // 32 channel blocks
#define NTHREADS        512
#define NWAVES          (NTHREADS / 32)
#define LINES_PER_ITER  (NWAVES * 4)   // 4 lines per wave per iteration
#define GRIDY           32
#define LINES_PER_WG    ((NLINES + GRIDY - 1) / GRIDY)   // 1563

#define FEAT_LDS_FLOATS  (CB * HW_)        // 65536 floats = 256 KB, layout [pixel][c]
#define STAGE_LDS_FLOATS (NWAVES * 128)    // 8 KB per-wave output transpose staging
#define SMEM_BYTES ((FEAT_LDS_FLOATS + STAGE_LDS_FLOATS) * 4)  // 270336 <= 320 KB

// Async global->LDS copy of 4 bytes (CDNA5 GLOBAL_LOAD_ASYNC_TO_LDS_B32, GV mode).
// VDST = per-lane LDS byte address, VADDR = per-lane 64-bit global address.
__device__ __forceinline__ void async_copy_b32(uint32_t lds_byte_off, const float* gsrc) {
    asm volatile("global_load_async_to_lds_b32 %0, %1, off"
                 :: "v"(lds_byte_off), "v"(gsrc)
                 : "memory");
}

__device__ __forceinline__ void wait_asynccnt0() {
    asm volatile("s_wait_asynccnt 0x0" ::: "memory");
}

__global__ __launch_bounds__(NTHREADS, 1)
void line_pool_kernel(const float* __restrict__ feat,     // [128][128][128]
                      const float* __restrict__ lines,    // [50000][4]
                      float* __restrict__ out)            // [50000][1024]
{
    extern __shared__ float smem[];                 // features + staging
    float* stage = smem + FEAT_LDS_FLOATS;

    const int tid = threadIdx.x;
    const int cb  = blockIdx.x;                     // channel block 0..31
    const float* featblk = feat + (size_t)cb * CB * HW_;

    // ---- stage 4 channels into LDS, channel-interleaved: lds[pix*4 + c] ----
    // Consecutive threads read consecutive pixels of one channel (coalesced
    // 128B global segments), scattered 16B-stride writes land in LDS.
    const uint32_t smem_base = (uint32_t)(uintptr_t)smem;   // low 32 bits = LDS offset
    for (int q = tid; q < FEAT_LDS_FLOATS; q += NTHREADS) {
        const int c   = q >> 14;            // q / HW_
        const int pix = q & (HW_ - 1);
        async_copy_b32(smem_base + (uint32_t)((pix * CB + c) * 4), featblk + q);
    }
    wait_asynccnt0();
    __syncthreads();

    const float4* __restrict__ smem4  = (const float4*)smem;   // [pixel] -> 4 channels
    const float4* __restrict__ lines4 = (const float4*)lines;

    const int wave = tid >> 5;
    const int lane = tid & 31;
    const int ll   = lane >> 3;        // line-in-wave   0..3
    const int f    = lane & 7;         // final bin      0..7

    const int wgStart = blockIdx.y * LINES_PER_WG;
    const int wgEnd   = (wgStart + LINES_PER_WG < NLINES) ? (wgStart + LINES_PER_WG) : NLINES;

    float* wstage = stage + wave * 128;     // 4 lines x 32 floats, per-wave private

    for (int base = wgStart + wave * 4; base < wgEnd; base += LINES_PER_ITER) {
        const int  line  = base + ll;
        const bool alive = (line < wgEnd);
        const float4 L = alive ? lines4[line] : make_float4(0.f, 0.f, 0.f, 0.f);

        float4 amax = make_float4(0.f, 0.f, 0.f, 0.f);
        #pragma unroll
        for (int s = 0; s < 4; ++s) {
            const float t  = (float)(f * 4 + s) * (1.0f / 31.0f);
            const float px = L.x * t + L.z * (1.0f - t) - 0.5f;
            const float py = L.y * t + L.w * (1.0f - t) - 0.5f;

            const float px0 = fminf(fmaxf(floorf(px), 0.0f), 127.0f);
            const float py0 = fminf(fmaxf(floorf(py), 0.0f), 127.0f);
            const float px1 = fminf(px0 + 1.0f, 127.0f);
            const float py1 = fminf(py0 + 1.0f, 127.0f);

            const int ix0 = (int)px0, iy0 = (int)py0;
            const int ix1 = (int)px1, iy1 = (int)py1;

            const float wx0 = px1 - px, wx1 = px - px0;
            const float wy0 = py1 - py, wy1 = py - py0;
            const float w00 = wy0 * wx0, w01 = wy0 * wx1;
            const float w10 = wy1 * wx0, w11 = wy1 * wx1;

            // one ds_load_b128 per corner fetches all 4 channels (no bank conflict)
            const float4 v00 = smem4[iy0 * WIMG + ix0];
            const float4 v01 = smem4[iy0 * WIMG + ix1];
            const float4 v10 = smem4[iy1 * WIMG + ix0];
            const float4 v11 = smem4[iy1 * WIMG + ix1];

            float4 r;
            r.x = v00.x * w00 + v01.x * w01 + v10.x * w10 + v11.x * w11;
            r.y = v00.y * w00 + v01.y * w01 + v10.y * w10 + v11.y * w11;
            r.z = v00.z * w00 + v01.z * w01 + v10.z * w10 + v11.z * w11;
            r.w = v00.w * w00 + v01.w * w01 + v10.w * w10 + v11.w * w11;

            if (s == 0) {
                amax = r;
            } else {
                amax.x = fmaxf(amax.x, r.x);
                amax.y = fmaxf(amax.y, r.y);
                amax.z = fmaxf(amax.z, r.z);
                amax.w = fmaxf(amax.w, r.w);
            }
        }

        // per-wave transpose through LDS so stores are full 128B cachelines
        wstage[ll * 32 + 0 * 8 + f] = amax.x;
        wstage[ll * 32 + 1 * 8 + f] = amax.y;
        wstage[ll * 32 + 2 * 8 + f] = amax.z;
        wstage[ll * 32 + 3 * 8 + f] = amax.w;

        // same-wave LDS ops are in-order; compiler inserts s_wait_dscnt for the RAW
        const float4 o = ((const float4*)wstage)[lane];   // lane -> line (lane>>3), chunk (lane&7)
        const int rline = base + (lane >> 3);
        if (rline < wgEnd) {
            // out[rline, cb*32 .. cb*32+31] is contiguous: (c*8+f) with c in block
            float4* dst = (float4*)(out + (size_t)rline * (CTOT * NFIN) + cb * 32 + (lane & 7) * 4);
            *dst = o;
        }
    }
}

extern "C" void kernel_launch(void* const* d_in, const int* in_sizes, int n_in,
                              void* d_out, int out_size, void* d_ws, size_t ws_size,
                              hipStream_t stream) {
    (void)in_sizes; (void)n_in; (void)d_ws; (void)ws_size; (void)out_size;
    const float* feat  = (const float*)d_in[0];   // (128,128,128) fp32
    const float* lines = (const float*)d_in[1];   // (50000,4) fp32
    float* out = (float*)d_out;                   // (50000,1024) fp32

    // allow >64KB dynamic LDS (gfx1250: up to 320KB per workgroup)
    hipFuncSetAttribute(reinterpret_cast<const void*>(line_pool_kernel),
                        hipFuncAttributeMaxDynamicSharedMemorySize, SMEM_BYTES);

    dim3 grid(NCB, GRIDY);   // 32 channel blocks x 32 line chunks
    line_pool_kernel<<<grid, NTHREADS, SMEM_BYTES, stream>>>(feat, lines, out);
}